// MetaLearner_8890582303116
// MI455X (gfx1250) — compile-verified
//
#include <hip/hip_runtime.h>
#include <hip/hip_bf16.h>
#include <math.h>

// CTC forward loss, fused log-softmax + alpha recursion.
// T=1024, N=128, C=256, S=64, L=129. One wave32 per batch item.
// Async global->LDS prefetch pipeline (CDNA5 GLOBAL_LOAD_ASYNC_TO_LDS_B128,
// tracked by ASYNCcnt) hides HBM latency behind the sequential DP.

#define T_DIM 1024
#define N_DIM 128
#define C_DIM 256
#define S_DIM 64
#define L_DIM 129          // 2*S+1
#define NEGV (-1e30f)
#define PIPE 8             // async prefetch depth (rows of 1KB); 8KB LDS/block
#define NJ 5               // states per lane: s = lane + 32*j

__device__ __forceinline__ float logaddexp3f(float x, float y, float z) {
    float m = fmaxf(x, fmaxf(y, z));
    return m + __logf(__expf(x - m) + __expf(y - m) + __expf(z - m));
}

__device__ __forceinline__ void async_row_load(unsigned int lds_addr,
                                               unsigned long long gaddr) {
    // Each lane moves 16B; two issues cover a 1KB row (offset applies to both
    // the LDS and global addresses per the CDNA5 async pseudocode).
    asm volatile(
        "global_load_async_to_lds_b128 %0, %1, off\n\t"
        "global_load_async_to_lds_b128 %0, %1, off offset:512"
        :: "v"(lds_addr), "v"(gaddr) : "memory");
}

__global__ __launch_bounds__(32) void ctc_alpha_kernel(
    const float* __restrict__ logits, const int* __restrict__ y,
    const int* __restrict__ xlens, const int* __restrict__ ylens,
    float* __restrict__ per_n) {

    __shared__ __align__(16) float rows[PIPE][C_DIM];

    const int n    = blockIdx.x;
    const int lane = threadIdx.x;
    const int xl   = xlens[n];
    const int yl   = ylens[n];

    // ---- per-lane extended-label metadata for states s = lane + 32*j ----
    int  elab[NJ];
    bool allow[NJ];
    bool valid[NJ];
#pragma unroll
    for (int j = 0; j < NJ; ++j) {
        const int s = lane + 32 * j;
        valid[j] = (s < L_DIM);
        int  e = 0;       // blank
        bool a = false;
        if (valid[j] && (s & 1)) {
            const int idx = (s - 1) >> 1;         // idx in [0, S)
            e = y[n * S_DIM + idx];               // labels in [1, C)
            if (s >= 3) a = (e != y[n * S_DIM + idx - 1]);
        }
        elab[j]  = e;
        allow[j] = a;
    }

    // ---- async prefetch pipeline prologue: rows 0..PIPE-1 ----
    const unsigned long long strideT =
        (unsigned long long)N_DIM * C_DIM * sizeof(float);
    unsigned long long g =
        (unsigned long long)(const void*)(logits + (size_t)n * C_DIM) +
        (unsigned long long)lane * 16ull;
    const unsigned int lds_base =
        (unsigned int)(unsigned long long)(const void*)&rows[0][0];
#pragma unroll
    for (int r = 0; r < PIPE; ++r) {
        async_row_load(lds_base + (unsigned)r * 1024u + (unsigned)lane * 16u, g);
        g += strideT;
    }
    unsigned long long gnext = g;

    float alpha[NJ];
#pragma unroll
    for (int j = 0; j < NJ; ++j) alpha[j] = NEGV;

    for (int t = 0; t < T_DIM; ++t) {
        const int buf = t & (PIPE - 1);

        // Wait until this row's two async loads landed (in-order completion).
        if (t <= T_DIM - PIPE) {
            asm volatile("s_wait_asynccnt 0xE" ::: "memory");
        } else {
            asm volatile("s_wait_asynccnt 0x0" ::: "memory");
        }

        // ---- lse = logsumexp over the 256-class row (in-wave reduction) ----
        const float4* r4 = (const float4*)(&rows[buf][0]);
        const float4 v0 = r4[lane * 2];
        const float4 v1 = r4[lane * 2 + 1];
        float m = fmaxf(fmaxf(fmaxf(v0.x, v0.y), fmaxf(v0.z, v0.w)),
                        fmaxf(fmaxf(v1.x, v1.y), fmaxf(v1.z, v1.w)));
#pragma unroll
        for (int o = 16; o >= 1; o >>= 1) m = fmaxf(m, __shfl_xor(m, o));
        float se = __expf(v0.x - m) + __expf(v0.y - m) +
                   __expf(v0.z - m) + __expf(v0.w - m) +
                   __expf(v1.x - m) + __expf(v1.y - m) +
                   __expf(v1.z - m) + __expf(v1.w - m);
#pragma unroll
        for (int o = 16; o >= 1; o >>= 1) se += __shfl_xor(se, o);
        const float lse = m + __logf(se);

        // ---- gather log-probs of extended labels ----
        float lp[NJ];
#pragma unroll
        for (int j = 0; j < NJ; ++j)
            lp[j] = valid[j] ? (rows[buf][elab[j]] - lse) : NEGV;

        if (t == 0) {
#pragma unroll
            for (int j = 0; j < NJ; ++j) {
                const int s = lane + 32 * j;
                alpha[j] = (valid[j] && s < 2) ? lp[j] : NEGV;
            }
        } else {
            // Shifted alphas: a1 = alpha[s-1], a2 = alpha[s-2] (skip-masked).
            float a1[NJ], a2[NJ];
#pragma unroll
            for (int j = 0; j < NJ; ++j) {
                const float up1 = __shfl_up(alpha[j], 1);
                const float up2 = __shfl_up(alpha[j], 2);
                const float p31 = (j > 0) ? __shfl(alpha[j - 1], 31) : NEGV;
                const float p30 = (j > 0) ? __shfl(alpha[j - 1], 30 + lane) : NEGV;
                a1[j] = (lane == 0) ? p31 : up1;
                a2[j] = (lane < 2) ? p30 : up2;
            }
            const bool upd = (t < xl);
#pragma unroll
            for (int j = 0; j < NJ; ++j) {
                const float a2v = allow[j] ? a2[j] : NEGV;
                const float nv  = lp[j] + logaddexp3f(alpha[j], a1[j], a2v);
                if (upd) alpha[j] = valid[j] ? nv : NEGV;
            }
        }

        // Ensure our LDS reads of this buffer are done before refilling it.
        asm volatile("s_wait_dscnt 0x0" ::: "memory");
        if (t + PIPE < T_DIM) {
            async_row_load(lds_base + (unsigned)buf * 1024u +
                               (unsigned)lane * 16u,
                           gnext);
            gnext += strideT;
        }
    }

    // ---- finalize: nll = -logaddexp(alpha[2*yl], alpha[2*yl-1]) ----
    const int end = 2 * yl;                // in [32, 128]
    const int je = end >> 5, le = end & 31;
    const int jb = (end - 1) >> 5, lb_ = (end - 1) & 31;
    float la = NEGV, lb = NEGV;
#pragma unroll
    for (int j = 0; j < NJ; ++j) {
        const float va = __shfl(alpha[j], le);
        const float vb = __shfl(alpha[j], lb_);
        if (j == je) la = va;
        if (j == jb) lb = vb;
    }
    const float mm = fmaxf(la, lb);
    float nll = -(mm + __logf(__expf(la - mm) + __expf(lb - mm)));
    const bool ok = isfinite(nll) && (nll < 1e29f);   // zero_infinity
    nll = ok ? nll : 0.0f;
    if (lane == 0) per_n[n] = nll / (float)yl;
}

// Deterministic mean over N (fixed-order tree reduction, no atomics).
__global__ __launch_bounds__(32) void ctc_reduce_kernel(
    const float* __restrict__ per_n, float* __restrict__ out) {
    const int lane = threadIdx.x;
    float s = 0.0f;
    for (int i = lane; i < N_DIM; i += 32) s += per_n[i];
#pragma unroll
    for (int o = 16; o >= 1; o >>= 1) s += __shfl_xor(s, o);
    if (lane == 0) out[0] = s / (float)N_DIM;
}

extern "C" void kernel_launch(void* const* d_in, const int* in_sizes, int n_in,
                              void* d_out, int out_size, void* d_ws, size_t ws_size,
                              hipStream_t stream) {
    const float* logits = (const float*)d_in[0];
    const int*   y      = (const int*)d_in[1];
    const int*   xlens  = (const int*)d_in[2];
    const int*   ylens  = (const int*)d_in[3];
    float* per_n = (float*)d_ws;          // N floats of scratch
    float* out   = (float*)d_out;

    ctc_alpha_kernel<<<N_DIM, 32, 0, stream>>>(logits, y, xlens, ylens, per_n);
    ctc_reduce_kernel<<<1, 32, 0, stream>>>(per_n, out);
}